// CustomFlashAttention_22694607192551
// MI455X (gfx1250) — compile-verified
//
#include <hip/hip_runtime.h>
#include <hip/hip_bf16.h>

// ---------------- problem constants (match reference setup_inputs) ----------
constexpr int B      = 16;
constexpr int H      = 2048;
constexpr int NH     = 16;
constexpr int D      = 128;            // H / NH
constexpr int MAXL   = 4096;
constexpr int CHUNK  = 512;            // flash-decode L split
constexpr int NCHUNK = MAXL / CHUNK;   // 8
constexpr int KC     = 4;              // K-split for WMMA projections
constexpr float SCALE = 0.08838834764831845f; // 1/sqrt(128)

typedef __attribute__((ext_vector_type(2))) float v2f;
typedef __attribute__((ext_vector_type(8))) float v8f;

// ---------------------------------------------------------------------------
// Projection partial GEMM: part[kc] = X(16xH) @ W^T over K-range of this kc.
// One wave (32 threads) per 16x16 output tile, V_WMMA_F32_16X16X4_F32.
// ISA 7.12.2 layouts (wave32):
//   A 16x4 f32 : lanes 0-15 = M, VGPR0/1 = K0/K1 ; lanes 16-31 = K2/K3
//   B 4x16 f32 : lanes 0-15 = N, VGPR0/1 = K0/K1 ; lanes 16-31 = K2/K3
//   C 16x16 f32: VGPR r -> M = (lane>=16?8:0)+r, N = lane&15
// B[k][n] = W[n][k] (since we need X @ W^T), so both A and B read pairs of
// consecutive K elements -> 8-byte global_load_b64 per operand per lane.
// ---------------------------------------------------------------------------
__global__ __launch_bounds__(32)
void proj16_wmma_part(const float* __restrict__ X, const float* __restrict__ W,
                      float* __restrict__ part) {
  const int lane = threadIdx.x;
  const int hi   = lane >> 4;        // 0/1 -> K pair select
  const int lo   = lane & 15;        // M for A, N for B
  const int nt   = blockIdx.x;       // N tile (H/16 tiles)
  const int kc   = blockIdx.y;       // K chunk
  const int n    = nt * 16 + lo;

  const float* xrow = X + (size_t)lo * H;   // A row m = lo
  const float* wrow = W + (size_t)n  * H;   // B column n = W row n
  const int k0 = kc * (H / KC);
  const int k1 = k0 + (H / KC);

  v8f c0 = {}, c1 = {}, c2 = {}, c3 = {};   // 4 chains for WMMA ILP
  for (int k = k0; k < k1; k += 16) {
    const v2f a0 = *(const v2f*)(xrow + k +  0 + 2 * hi);
    const v2f b0 = *(const v2f*)(wrow + k +  0 + 2 * hi);
    const v2f a1 = *(const v2f*)(xrow + k +  4 + 2 * hi);
    const v2f b1 = *(const v2f*)(wrow + k +  4 + 2 * hi);
    const v2f a2 = *(const v2f*)(xrow + k +  8 + 2 * hi);
    const v2f b2 = *(const v2f*)(wrow + k +  8 + 2 * hi);
    const v2f a3 = *(const v2f*)(xrow + k + 12 + 2 * hi);
    const v2f b3 = *(const v2f*)(wrow + k + 12 + 2 * hi);
    c0 = __builtin_amdgcn_wmma_f32_16x16x4_f32(false, a0, false, b0, (short)0, c0, false, false);
    c1 = __builtin_amdgcn_wmma_f32_16x16x4_f32(false, a1, false, b1, (short)0, c1, false, false);
    c2 = __builtin_amdgcn_wmma_f32_16x16x4_f32(false, a2, false, b2, (short)0, c2, false, false);
    c3 = __builtin_amdgcn_wmma_f32_16x16x4_f32(false, a3, false, b3, (short)0, c3, false, false);
  }
  const v8f c = (c0 + c1) + (c2 + c3);

  float* prow = part + (size_t)kc * (B * H);
#pragma unroll
  for (int r = 0; r < 8; ++r)
    prow[(size_t)(hi * 8 + r) * H + nt * 16 + lo] = c[r];
}

// Deterministic K-split reduction + bias add.
__global__ __launch_bounds__(256)
void proj_reduce(const float* __restrict__ part, const float* __restrict__ bias,
                 float* __restrict__ out) {
  const int idx = blockIdx.x * 256 + threadIdx.x;   // < B*H
  const int n   = idx & (H - 1);
  float s = bias[n];
#pragma unroll
  for (int kc = 0; kc < KC; ++kc) s += part[(size_t)kc * (B * H) + idx];
  out[idx] = s;
}

// ---------------------------------------------------------------------------
// Flash-decode partial: one (batch,head) x one L-chunk per CTA (128 threads).
// K/V caches are NOT mutated; the freshly projected row is substituted at
// j == current_pos. Writes (acc[128], m, l) partial per chunk.
// ---------------------------------------------------------------------------
__global__ __launch_bounds__(128)
void attn_partial(const float* __restrict__ kcache, const float* __restrict__ vcache,
                  const float* __restrict__ qws, const float* __restrict__ kws,
                  const float* __restrict__ vws, const int* __restrict__ cpos_p,
                  float* __restrict__ fpart) {
  __shared__ float scores[CHUNK];
  __shared__ float qsh[D];
  __shared__ float red[4];

  const int bh    = blockIdx.x;             // b*NH + h
  const int chunk = blockIdx.y;
  const int b     = bh >> 4;
  const int h     = bh & 15;
  const int cpos  = *cpos_p;
  const int L     = cpos + 1;
  const int t     = threadIdx.x;            // 0..127 (owns d index t)
  const int wv    = t >> 5;
  const int lane  = t & 31;
  const float NEG_INF = -__builtin_inff();

  float* dst = fpart + (size_t)(bh * NCHUNK + chunk) * (D + 2);
  const int j0 = chunk * CHUNK;
  if (j0 >= L) {                            // fully-masked chunk: neutral partial
    dst[t] = 0.f;
    if (t == 0) { dst[D] = NEG_INF; dst[D + 1] = 0.f; }
    return;
  }

  qsh[t] = qws[(size_t)b * H + h * D + t];
  __syncthreads();

  const float* knew = kws + (size_t)b * H + h * D;
  const float* vnew = vws + (size_t)b * H + h * D;

  // Phase A: scores; one wave per row, float4-coalesced K loads, wave32 reduce.
  for (int jj = wv; jj < CHUNK; jj += 4) {
    const int j = j0 + jj;
    float s = NEG_INF;
    if (j < L) {
      const float* krow = (j == cpos) ? knew
                        : (kcache + ((size_t)b * MAXL + j) * H + h * D);
      const float4 kq = *(const float4*)(krow + lane * 4);
      const float4 qq = *(const float4*)(&qsh[lane * 4]);
      float dp = kq.x * qq.x + kq.y * qq.y + kq.z * qq.z + kq.w * qq.w;
#pragma unroll
      for (int off = 16; off > 0; off >>= 1)
        dp += __shfl_xor(dp, off, 32);
      s = dp * SCALE;
      // warm V row for phase B (global_prefetch_b8)
      const float* vrow = (j == cpos) ? vnew
                        : (vcache + ((size_t)b * MAXL + j) * H + h * D);
      __builtin_prefetch(vrow + lane * 4, 0, 0);
    }
    if (lane == 0) scores[jj] = s;
  }
  __syncthreads();

  // Phase B: chunk max (wave shuffles + LDS cross-wave).
  float lm = NEG_INF;
  for (int jj = t; jj < CHUNK; jj += 128) lm = fmaxf(lm, scores[jj]);
#pragma unroll
  for (int off = 16; off > 0; off >>= 1) lm = fmaxf(lm, __shfl_xor(lm, off, 32));
  if (lane == 0) red[wv] = lm;
  __syncthreads();
  const float m = fmaxf(fmaxf(red[0], red[1]), fmaxf(red[2], red[3]));

  // exp / sum / p*V accumulate; thread t owns d-index t (coalesced V loads).
  float acc = 0.f, lsum = 0.f;
  for (int jj = 0; jj < CHUNK; ++jj) {
    const int j = j0 + jj;
    const float p = __expf(scores[jj] - m);   // masked rows: exp(-inf)=0
    lsum += p;
    const float* vrow = (j == cpos) ? vnew
                      : (vcache + ((size_t)b * MAXL + j) * H + h * D);
    acc = fmaf(p, vrow[t], acc);
  }
  dst[t] = acc;
  if (t == 0) { dst[D] = m; dst[D + 1] = lsum; }
}

// Merge chunk partials -> attention output (16 x 2048).
__global__ __launch_bounds__(128)
void attn_combine(const float* __restrict__ fpart, float* __restrict__ attnws) {
  const int bh = blockIdx.x;
  const int t  = threadIdx.x;
  const int b  = bh >> 4;
  const int h  = bh & 15;
  const float* base = fpart + (size_t)bh * NCHUNK * (D + 2);
  float gm = -__builtin_inff();
#pragma unroll
  for (int c = 0; c < NCHUNK; ++c) gm = fmaxf(gm, base[c * (D + 2) + D]);
  float o = 0.f, s = 0.f;
#pragma unroll
  for (int c = 0; c < NCHUNK; ++c) {
    const float mc = base[c * (D + 2) + D];
    const float lc = base[c * (D + 2) + D + 1];
    const float w  = __expf(mc - gm);
    o = fmaf(base[c * (D + 2) + t], w, o);
    s = fmaf(lc, w, s);
  }
  attnws[(size_t)b * H + h * D + t] = o / s;
}

// ---------------------------------------------------------------------------
extern "C" void kernel_launch(void* const* d_in, const int* in_sizes, int n_in,
                              void* d_out, int out_size, void* d_ws, size_t ws_size,
                              hipStream_t stream) {
  const float* x   = (const float*)d_in[0];
  const float* kcc = (const float*)d_in[1];
  const float* vcc = (const float*)d_in[2];
  const float* w_q = (const float*)d_in[3];
  const float* w_k = (const float*)d_in[4];
  const float* w_v = (const float*)d_in[5];
  const float* w_o = (const float*)d_in[6];
  const float* b_q = (const float*)d_in[7];
  const float* b_k = (const float*)d_in[8];
  const float* b_v = (const float*)d_in[9];
  const float* b_o = (const float*)d_in[10];
  const int*  cpos = (const int*)d_in[11];
  float* out = (float*)d_out;

  // workspace layout (floats); reuse of `part` is stream-ordered safe
  float* ws    = (float*)d_ws;
  float* part  = ws;                        // KC*B*H   = 131072
  float* qws   = part + (size_t)KC * B * H; // B*H      =  32768
  float* kws   = qws  + (size_t)B * H;
  float* vws   = kws  + (size_t)B * H;
  float* aws   = vws  + (size_t)B * H;
  float* fpart = aws  + (size_t)B * H;      // B*NH*NCHUNK*(D+2) = 266240

  const dim3 pg(H / 16, KC), pb(32);
  const int rg = (B * H) / 256;

  proj16_wmma_part<<<pg, pb, 0, stream>>>(x, w_q, part);
  proj_reduce<<<rg, 256, 0, stream>>>(part, b_q, qws);
  proj16_wmma_part<<<pg, pb, 0, stream>>>(x, w_k, part);
  proj_reduce<<<rg, 256, 0, stream>>>(part, b_k, kws);
  proj16_wmma_part<<<pg, pb, 0, stream>>>(x, w_v, part);
  proj_reduce<<<rg, 256, 0, stream>>>(part, b_v, vws);

  attn_partial<<<dim3(B * NH, NCHUNK), 128, 0, stream>>>(kcc, vcc, qws, kws, vws, cpos, fpart);
  attn_combine<<<B * NH, 128, 0, stream>>>(fpart, aws);

  proj16_wmma_part<<<pg, pb, 0, stream>>>(aws, w_o, part);
  proj_reduce<<<rg, 256, 0, stream>>>(part, b_o, out);
}